// DGNN_40510131536131
// MI455X (gfx1250) — compile-verified
//
#include <hip/hip_runtime.h>
#include <hip/hip_bf16.h>
#include <math.h>

#define N_NODES 50000
#define E_EDGES 800000
#define D_FEAT  128
#define D_OUTC  40
#define D_OUTP  48          // D_OUTC padded to multiple of 16 for WMMA
#define BN_EPS  1e-5f

typedef __attribute__((ext_vector_type(2))) float v2f;
typedef __attribute__((ext_vector_type(8))) float v8f;

// Guaranteed-native FP32 atomic add (VGLOBAL GLOBAL_ATOMIC_ADD_F32, no return,
// device scope). Tracked by STOREcnt; s_endpgm implies wait-idle.
__device__ __forceinline__ void atomAddF(float* p, float v) {
    asm volatile("global_atomic_add_f32 %0, %1, off scope:SCOPE_DEV"
                 : : "v"(p), "v"(v) : "memory");
}

// ---------------------------------------------------------------------------
// GEMM: C[M x cols] = A[M x 128] * W[128 x ncp]   (f32 WMMA 16x16x4)
// ncp = cols padded to multiple of 16; W must be physically [128 x ncp]
// (zero-padded), so the inner loop is branch-free.
// grid.x = M/16; block = 256 = 8 waves; wave w owns N-tile n0 = 16*w.
// ---------------------------------------------------------------------------
__global__ __launch_bounds__(256) void gemm_wmma_f32(
    const float* __restrict__ A, const float* __restrict__ W,
    float* __restrict__ C, int ncp, int cols, int ldc)
{
    const int m0   = blockIdx.x * 16;
    const int wave = threadIdx.x >> 5;
    const int lane = threadIdx.x & 31;
    const int half = lane >> 4;     // 0: lanes 0-15, 1: lanes 16-31
    const int r    = lane & 15;
    const int n0   = wave * 16;
    if (n0 >= ncp) return;          // wave-uniform exit

    const float* arow = A + (size_t)(m0 + r) * D_FEAT;
    const float* wcol = W + (size_t)(2 * half) * ncp + (n0 + r);

    v8f acc = {0.f, 0.f, 0.f, 0.f, 0.f, 0.f, 0.f, 0.f};
#pragma unroll 4
    for (int k0 = 0; k0 < D_FEAT; k0 += 4) {
        const int kk = k0 + 2 * half;
        v2f a, b;
        // A 16x4: lane r = row m0+r; lanes 0-15 -> K=k0,k0+1 ; 16-31 -> K=k0+2,k0+3
        a.x = arow[kk];
        a.y = arow[kk + 1];
        // B 4x16 mirrored: lane r = col n0+r, same K pair
        b.x = wcol[(size_t)k0 * ncp];
        b.y = wcol[(size_t)k0 * ncp + ncp];
        acc = __builtin_amdgcn_wmma_f32_16x16x4_f32(
            /*neg_a=*/false, a, /*neg_b=*/false, b,
            /*c_mod=*/(short)0, acc, /*reuse_a=*/false, /*reuse_b=*/false);
    }
    // C/D layout: VGPR j, lanes 0-15 -> row m0+j ; lanes 16-31 -> row m0+8+j
    const int col = n0 + r;
    if (col < cols) {
#pragma unroll
        for (int j = 0; j < 8; ++j) {
            const int row = m0 + half * 8 + j;
            C[(size_t)row * ldc + col] = acc[j];
        }
    }
}

// Zero-pad W3 [128 x 40] -> W3p [128 x 48]
__global__ void pad_w3(const float* __restrict__ W3, float* __restrict__ W3p) {
    int i = blockIdx.x * blockDim.x + threadIdx.x;   // over 128*48
    if (i < D_FEAT * D_OUTP) {
        int k = i / D_OUTP, c = i - k * D_OUTP;
        W3p[i] = (c < D_OUTC) ? W3[k * D_OUTC + c] : 0.0f;
    }
}

// ---------------------------------------------------------------------------
// Degree / normalization (computed once; reused by all 3 layers)
// ---------------------------------------------------------------------------
__global__ void fill_ones(float* __restrict__ d, int n) {
    int i = blockIdx.x * blockDim.x + threadIdx.x;
    if (i < n) d[i] = 1.0f;                        // self-loop: deg starts at 1
}
__global__ void deg_count(const int* __restrict__ dst, float* __restrict__ deg, int e) {
    int i = blockIdx.x * blockDim.x + threadIdx.x;
    if (i < e) atomAddF(&deg[dst[i]], 1.0f);
}
__global__ void deg_rsqrt(float* __restrict__ d, int n) {
    int i = blockIdx.x * blockDim.x + threadIdx.x;
    if (i < n) d[i] = rsqrtf(d[i]);
}

// ---------------------------------------------------------------------------
// agg[i,f] = h[i,f] * dinv[i]^2 + bias[f]   (self-loop term + bias)
// ---------------------------------------------------------------------------
__global__ void self_init(const float* __restrict__ h, const float* __restrict__ dinv,
                          const float* __restrict__ bias, float* __restrict__ agg,
                          int cols, int total) {
    int i = blockIdx.x * blockDim.x + threadIdx.x;
    if (i < total) {
        int row = i / cols, f = i - row * cols;
        float di = dinv[row];
        agg[i] = h[i] * (di * di) + bias[f];
    }
}

// ---------------------------------------------------------------------------
// Edge scatter, D=128: one wave per edge; each lane moves a float4 (b128 load,
// 4 HW fp atomics). h and agg are L2-resident (25.6 MB each, 192 MB L2).
// ---------------------------------------------------------------------------
__global__ __launch_bounds__(256) void scatter_d128(
    const int* __restrict__ src, const int* __restrict__ dst,
    const float* __restrict__ dinv, const float* __restrict__ h,
    float* __restrict__ agg, int e_count)
{
    int e    = (int)((blockIdx.x * blockDim.x + threadIdx.x) >> 5);
    int lane = threadIdx.x & 31;
    if (e >= e_count) return;
    int s = src[e], d = dst[e];
    float coef = dinv[s] * dinv[d];
    const float4* hp = (const float4*)(h + (size_t)s * D_FEAT);
    float4 v = hp[lane];
    float* ap = agg + (size_t)d * D_FEAT + lane * 4;
    atomAddF(ap + 0, v.x * coef);
    atomAddF(ap + 1, v.y * coef);
    atomAddF(ap + 2, v.z * coef);
    atomAddF(ap + 3, v.w * coef);
}

// Edge scatter, D=40: wave per edge; lanes cover f=lane, lanes 0-7 also f=32+lane
__global__ __launch_bounds__(256) void scatter_d40(
    const int* __restrict__ src, const int* __restrict__ dst,
    const float* __restrict__ dinv, const float* __restrict__ h,
    float* __restrict__ agg, int e_count)
{
    int e    = (int)((blockIdx.x * blockDim.x + threadIdx.x) >> 5);
    int lane = threadIdx.x & 31;
    if (e >= e_count) return;
    int s = src[e], d = dst[e];
    float coef = dinv[s] * dinv[d];
    const float* hs = h + (size_t)s * D_OUTC;
    float* ap = agg + (size_t)d * D_OUTC;
    atomAddF(ap + lane, hs[lane] * coef);
    if (lane < 8) atomAddF(ap + 32 + lane, hs[32 + lane] * coef);
}

// ---------------------------------------------------------------------------
// out = relu((x - mean) * rsqrt(var+eps) * gamma + beta)
// ---------------------------------------------------------------------------
__global__ void bn_relu(const float* __restrict__ x,
                        const float* __restrict__ g, const float* __restrict__ be,
                        const float* __restrict__ m, const float* __restrict__ v,
                        float* __restrict__ out, int total) {
    int i = blockIdx.x * blockDim.x + threadIdx.x;
    if (i < total) {
        int f = i & (D_FEAT - 1);
        float y = (x[i] - m[f]) * rsqrtf(v[f] + BN_EPS) * g[f] + be[f];
        out[i] = fmaxf(y, 0.0f);
    }
}

// ---------------------------------------------------------------------------
// In-place log_softmax over rows of 40; one wave32 per row, butterfly reduce.
// ---------------------------------------------------------------------------
__global__ __launch_bounds__(256) void log_softmax40(float* __restrict__ out, int nrows) {
    int row  = (int)((blockIdx.x * blockDim.x + threadIdx.x) >> 5);
    int lane = threadIdx.x & 31;
    if (row >= nrows) return;
    float* p = out + (size_t)row * D_OUTC;
    float v0 = p[lane];
    float v1 = (lane < 8) ? p[32 + lane] : -1e30f;
    float mx = fmaxf(v0, v1);
#pragma unroll
    for (int off = 16; off > 0; off >>= 1) mx = fmaxf(mx, __shfl_xor(mx, off, 32));
    float s = expf(v0 - mx) + ((lane < 8) ? expf(v1 - mx) : 0.0f);
#pragma unroll
    for (int off = 16; off > 0; off >>= 1) s += __shfl_xor(s, off, 32);
    float lse = mx + logf(s);
    p[lane] = v0 - lse;
    if (lane < 8) p[32 + lane] = v1 - lse;
}

// ---------------------------------------------------------------------------
extern "C" void kernel_launch(void* const* d_in, const int* in_sizes, int n_in,
                              void* d_out, int out_size, void* d_ws, size_t ws_size,
                              hipStream_t stream) {
    const float* x   = (const float*)d_in[0];
    const int*   src = (const int*)  d_in[1];
    const int*   dst = (const int*)  d_in[2];
    const float* W1  = (const float*)d_in[3];
    const float* b1  = (const float*)d_in[4];
    const float* W2  = (const float*)d_in[5];
    const float* b2  = (const float*)d_in[6];
    const float* W3  = (const float*)d_in[7];
    const float* b3  = (const float*)d_in[8];
    const float* g1  = (const float*)d_in[9];
    const float* be1 = (const float*)d_in[10];
    const float* m1  = (const float*)d_in[11];
    const float* v1  = (const float*)d_in[12];
    const float* g2  = (const float*)d_in[13];
    const float* be2 = (const float*)d_in[14];
    const float* m2  = (const float*)d_in[15];
    const float* v2  = (const float*)d_in[16];
    float* out = (float*)d_out;

    // Workspace: dinv | W3p | bufA | bufB   (all L2-resident, 192MB L2)
    char* ws = (char*)d_ws;
    size_t off = 0;
    float* dinv = (float*)(ws + off); off += (((size_t)N_NODES * 4) + 255) & ~(size_t)255;
    float* W3p  = (float*)(ws + off); off += (size_t)D_FEAT * D_OUTP * 4;
    float* bufA = (float*)(ws + off); off += (size_t)N_NODES * D_FEAT * 4;
    float* bufB = (float*)(ws + off);

    const int TB = 256;
    const int nGemmBlk = N_NODES / 16;                 // 50000/16 = 3125 exact
    const int nElem128 = N_NODES * D_FEAT;
    const int nElem40  = N_NODES * D_OUTC;
    const int edgeWaveBlks = (E_EDGES * 32 + TB - 1) / TB;

    // degree -> dinv (once, reused by all layers) ; pad W3
    fill_ones<<<(N_NODES + TB - 1) / TB, TB, 0, stream>>>(dinv, N_NODES);
    deg_count<<<(E_EDGES + TB - 1) / TB, TB, 0, stream>>>(dst, dinv, E_EDGES);
    deg_rsqrt<<<(N_NODES + TB - 1) / TB, TB, 0, stream>>>(dinv, N_NODES);
    pad_w3<<<(D_FEAT * D_OUTP + TB - 1) / TB, TB, 0, stream>>>(W3, W3p);

    // ---- Layer 1 ----
    gemm_wmma_f32<<<nGemmBlk, TB, 0, stream>>>(x, W1, bufA, D_FEAT, D_FEAT, D_FEAT);
    self_init<<<(nElem128 + TB - 1) / TB, TB, 0, stream>>>(bufA, dinv, b1, bufB, D_FEAT, nElem128);
    scatter_d128<<<edgeWaveBlks, TB, 0, stream>>>(src, dst, dinv, bufA, bufB, E_EDGES);
    bn_relu<<<(nElem128 + TB - 1) / TB, TB, 0, stream>>>(bufB, g1, be1, m1, v1, bufA, nElem128);

    // ---- Layer 2 ----
    gemm_wmma_f32<<<nGemmBlk, TB, 0, stream>>>(bufA, W2, bufB, D_FEAT, D_FEAT, D_FEAT);
    self_init<<<(nElem128 + TB - 1) / TB, TB, 0, stream>>>(bufB, dinv, b2, bufA, D_FEAT, nElem128);
    scatter_d128<<<edgeWaveBlks, TB, 0, stream>>>(src, dst, dinv, bufB, bufA, E_EDGES);
    bn_relu<<<(nElem128 + TB - 1) / TB, TB, 0, stream>>>(bufA, g2, be2, m2, v2, bufB, nElem128);

    // ---- Layer 3 (agg directly into d_out, then in-place log_softmax) ----
    gemm_wmma_f32<<<nGemmBlk, TB, 0, stream>>>(bufB, W3p, bufA, D_OUTP, D_OUTC, D_OUTC);
    self_init<<<(nElem40 + TB - 1) / TB, TB, 0, stream>>>(bufA, dinv, b3, out, D_OUTC, nElem40);
    scatter_d40<<<edgeWaveBlks, TB, 0, stream>>>(src, dst, dinv, bufA, out, E_EDGES);
    log_softmax40<<<(N_NODES * 32 + TB - 1) / TB, TB, 0, stream>>>(out, N_NODES);
}